// Net_55980603736260
// MI455X (gfx1250) — compile-verified
//
#include <hip/hip_runtime.h>
#include <hip/hip_bf16.h>

// ---------------------------------------------------------------------------
// CDNA5 (gfx1250) implementation. All three GEMMs run on V_WMMA_F32_16X16X4_F32
// (native f32 matrix op, wave32). Conv is expressed as a unified im2col GEMM
// with a 5-tap window covering all four SAME-padded kernel sizes.
// Register-blocked (13/5/4 accumulator tiles per wave) + LDS-staged B panels
// (double buffered) to keep L2 traffic ~8x below the naive per-wave streaming.
// ---------------------------------------------------------------------------

typedef float v2f __attribute__((ext_vector_type(2)));
typedef float v8f __attribute__((ext_vector_type(8)));

#define B_    512
#define L_    128
#define E_    128
#define H_    194
#define HP    208     // H padded to 13*16
#define TCLS  68
#define TP    80      // T padded to 5*16
#define ACLS  56
#define AP    64      // A padded to 4*16
#define NC_   16
#define NT_   8
#define KKC   160     // conv K = 640 = E*5 taps, /4
#define KKT   52      // trigger K = 208, /4
#define KKA   104     // argument K = 416 = 2*HP, /4

// workspace layout (float offsets)
#define OFF_WPACK 0
#define SZ_WPACK  (KKC*HP*4)                // 133120
#define OFF_WTP   (OFF_WPACK + SZ_WPACK)
#define SZ_WTP    (KKT*TP*4)                // 16640
#define OFF_WAP   (OFF_WTP + SZ_WTP)
#define SZ_WAP    (KKA*AP*4)                // 26624
#define OFF_BIAS  (OFF_WAP + SZ_WAP)
#define SZ_BIAS   HP
#define OFF_TRGS  (OFF_BIAS + SZ_BIAS)
#define SZ_TRGS   (B_*NT_*HP)               // 851968
#define OFF_ENTS  (OFF_TRGS + SZ_TRGS)
#define SZ_ENTS   (B_*NC_*HP)               // 1703936
// total ~10.9 MB of scratch

#define XS_STRIDE 130           // 128 cols + pad (bank-conflict avoidance)
#define CS_STRIDE 210           // 208 cols + pad
#define PKK       8             // conv K-steps per staged weight panel
#define NPAN      (KKC/PKK)     // 20 panels
#define PANEL     (PKK*HP*4)    // 6656 floats per panel
#define SMEM_FUSED ((132*XS_STRIDE + 128*CS_STRIDE + 2*PANEL) * sizeof(float)) // 229,408 B

#define SSTR      210           // span-row stride in LDS (bank-conflict pad)
#define SMEM_ARG  ((24*SSTR + SZ_WAP) * sizeof(float))                         // 126,656 B

static __device__ __forceinline__ v8f wmma_f32(v2f a, v2f b, v8f c) {
    return __builtin_amdgcn_wmma_f32_16x16x4_f32(
        /*neg_a=*/false, a, /*neg_b=*/false, b,
        /*c_mod=*/(short)0, c, /*reuse_a=*/false, /*reuse_b=*/false);
}

// ---------------------------------------------------------------------------
// Kernel 0: pack weights into WMMA-friendly [K/4][N][4] layouts (B-fragment =
// one aligned float2 per lane), embed all 4 conv branches into a single
// 640x208 matrix at their SAME-padding tap offsets (k=2:p0, 3:p1, 4:p1, 5:p2).
// ---------------------------------------------------------------------------
__global__ void pack_kernel(const float* __restrict__ w0, const float* __restrict__ w1,
                            const float* __restrict__ w2, const float* __restrict__ w3,
                            const float* __restrict__ b0, const float* __restrict__ b1,
                            const float* __restrict__ b2, const float* __restrict__ b3,
                            const float* __restrict__ wt, const float* __restrict__ wa,
                            float* __restrict__ ws) {
    int idx = blockIdx.x * blockDim.x + threadIdx.x;
    const int total = SZ_WPACK + SZ_WTP + SZ_WAP + SZ_BIAS;
    if (idx >= total) return;

    if (idx < SZ_WPACK) {
        int j = idx & 3, rest = idx >> 2;
        int n = rest % HP, kk = rest / HP;
        int k = kk * 4 + j;
        int tap = k >> 7, e = k & 127;
        float v = 0.f;
        int c = 0, ksz = 0, p = 0; const float* w = nullptr;
        if (n < 50)       { c = n;       ksz = 2; p = 0; w = w0; }
        else if (n < 99)  { c = n - 50;  ksz = 3; p = 1; w = w1; }
        else if (n < 147) { c = n - 99;  ksz = 4; p = 1; w = w2; }
        else if (n < 194) { c = n - 147; ksz = 5; p = 2; w = w3; }
        if (w) {
            int dk = tap - 2 + p;                 // window pos -> conv tap
            if (dk >= 0 && dk < ksz) v = w[(c * E_ + e) * ksz + dk];
        }
        ws[OFF_WPACK + idx] = v;
    } else if (idx < SZ_WPACK + SZ_WTP) {
        int li = idx - SZ_WPACK;
        int j = li & 3, rest = li >> 2;
        int n = rest % TP, kk = rest / TP;
        int k = kk * 4 + j;
        float v = 0.f;
        if (k < H_ && n < TCLS) v = wt[k * TCLS + n];
        ws[OFF_WTP + li] = v;
    } else if (idx < SZ_WPACK + SZ_WTP + SZ_WAP) {
        int li = idx - SZ_WPACK - SZ_WTP;
        int j = li & 3, rest = li >> 2;
        int n = rest % AP, kk = rest / AP;
        int k = kk * 4 + j;
        float v = 0.f;
        if (n < ACLS) {
            if (k < H_)                      v = wa[k * ACLS + n];              // trigger half
            else if (k >= HP && k < HP + H_) v = wa[(k - HP + H_) * ACLS + n];  // entity half
        }
        ws[OFF_WAP + li] = v;
    } else {
        int n = idx - SZ_WPACK - SZ_WTP - SZ_WAP;
        float v = 0.f;
        if (n < 50)       v = b0[n];
        else if (n < 99)  v = b1[n - 50];
        else if (n < 147) v = b2[n - 99];
        else if (n < 194) v = b3[n - 147];
        ws[OFF_BIAS + n] = v;
    }
}

// ---------------------------------------------------------------------------
// Kernel 1 (fused, one batch per block, 8 waves, 224KB LDS -> 1 WG/WGP):
//   gather emb rows -> LDS xs (2-row zero halo)
//   conv GEMM   (WMMA f32, K=640): 13 accumulators/wave, double-buffered
//               8-K-step weight panels staged in LDS -> bias+ReLU -> LDS cs
//   trigger GEMM (WMMA f32, K=208): wt staged into the dead xs region
//   span means (<=7-row sums from LDS) -> trgS/entS scratch
// ---------------------------------------------------------------------------
__global__ __launch_bounds__(256) void fused_kernel(
    const int* __restrict__ tokens,
    const int* __restrict__ cstart, const int* __restrict__ clen,
    const int* __restrict__ tstart, const int* __restrict__ tlen,
    const float* __restrict__ emb, const float* __restrict__ bt,
    const float* __restrict__ ws,
    float* __restrict__ trgS, float* __restrict__ entS,
    float* __restrict__ out_trig) {
    extern __shared__ float smem[];
    float* xs  = smem;                               // [132][XS_STRIDE]
    float* cs  = smem + 132 * XS_STRIDE;             // [128][CS_STRIDE]
    float* wp0 = cs + 128 * CS_STRIDE;               // panel buffer 0
    float* wp1 = wp0 + PANEL;                        // panel buffer 1

    const int b    = blockIdx.x;
    const int tid  = threadIdx.x;
    const int lane = tid & 31;
    const int wave = tid >> 5;              // 0..7 -> M-tile row
    const int col16 = lane & 15;
    const int koff  = (lane >> 4) << 1;     // lanes 0-15: K+0/K+1, 16-31: K+2/K+3
    const int rofs  = (lane >> 4) << 3;     // C/D: upper lanes hold M = r+8

    // zero the 2-row halos (rows 0,1,130,131)
    for (int i = tid; i < 2 * XS_STRIDE; i += 256) {
        xs[i] = 0.f;
        xs[130 * XS_STRIDE + i] = 0.f;
    }
    // embedding gather: 2 threads per sequence row, 64 floats each (float4)
    {
        int row  = tid >> 1;
        int half = (tid & 1) * 64;
        int t = tokens[b * L_ + row];
        const float4* src = (const float4*)(emb + (long)t * E_ + half);
        float* dst = &xs[(row + 2) * XS_STRIDE + half];
#pragma unroll
        for (int i = 0; i < 16; ++i) {
            float4 v = src[i];
            dst[4*i+0] = v.x; dst[4*i+1] = v.y; dst[4*i+2] = v.z; dst[4*i+3] = v.w;
        }
    }
    const float* Wpack = ws + OFF_WPACK;
    const float* bias  = ws + OFF_BIAS;

    // stage first weight panel
    for (int i = tid; i < PANEL; i += 256) wp0[i] = Wpack[i];
    __syncthreads();

    // ---- conv GEMM: cs[l][h] = relu(sum_k xs_win[l][k] * Wbig[k][h] + b[h])
    const v8f vzero = {0.f,0.f,0.f,0.f,0.f,0.f,0.f,0.f};
    v8f acc[13];
#pragma unroll
    for (int n = 0; n < 13; ++n) acc[n] = vzero;

    for (int kp = 0; kp < NPAN; ++kp) {
        float* cur = (kp & 1) ? wp1 : wp0;
        float* nxt = (kp & 1) ? wp0 : wp1;
        if (kp + 1 < NPAN) {                       // prefetch next panel
            const float* src = Wpack + (kp + 1) * PANEL;
            for (int i = tid; i < PANEL; i += 256) nxt[i] = src[i];
        }
#pragma unroll 2
        for (int k2 = 0; k2 < PKK; ++k2) {
            int kk  = kp * PKK + k2;
            int k   = kk << 2;
            int tap = k >> 7;
            int e   = (k & 127) + koff;            // never crosses a tap (E%4==0)
            int arow = wave * 16 + col16 + tap;    // halo offset folded in
            v2f a; a.x = xs[arow * XS_STRIDE + e];
                   a.y = xs[arow * XS_STRIDE + e + 1];
            const float* pb = cur + ((k2 * HP + col16) << 2) + koff;
#pragma unroll
            for (int n = 0; n < 13; ++n) {
                v2f bb = *(const v2f*)(pb + ((n * 16) << 2));
                acc[n] = wmma_f32(a, bb, acc[n]);
            }
        }
        __syncthreads();   // all reads of cur done; nxt fully written
    }
    // write conv results (bias + ReLU) into LDS cs  (own-wave rows only)
#pragma unroll
    for (int n = 0; n < 13; ++n) {
#pragma unroll
        for (int r = 0; r < 8; ++r) {
            int row = wave * 16 + r + rofs;
            int col = n * 16 + col16;
            float v = acc[n][r] + bias[col];
            cs[row * CS_STRIDE + col] = v > 0.f ? v : 0.f;
        }
    }
    // xs is dead now: recycle its LDS region for the trigger weight matrix
    {
        const float* wtpG = ws + OFF_WTP;
        for (int i = tid; i < SZ_WTP; i += 256) xs[i] = wtpG[i];
    }
    __syncthreads();       // cs complete (all waves) + wtp staged

    // ---- trigger GEMM: out[l][t] = cs[l][:] @ wt + bt  (A and B from LDS)
    {
        v8f tacc[5];
#pragma unroll
        for (int n = 0; n < 5; ++n) tacc[n] = vzero;
#pragma unroll 2
        for (int kk = 0; kk < KKT; ++kk) {
            int k = (kk << 2) + koff;
            int arow = wave * 16 + col16;
            v2f a; a.x = cs[arow * CS_STRIDE + k];
                   a.y = cs[arow * CS_STRIDE + k + 1];
            const float* pb = xs + ((kk * TP + col16) << 2) + koff;
#pragma unroll
            for (int n = 0; n < 5; ++n) {
                v2f bb = *(const v2f*)(pb + ((n * 16) << 2));
                tacc[n] = wmma_f32(a, bb, tacc[n]);
            }
        }
#pragma unroll
        for (int n = 0; n < 5; ++n) {
#pragma unroll
            for (int r = 0; r < 8; ++r) {
                int row = wave * 16 + r + rofs;
                int col = n * 16 + col16;
                if (col < TCLS)
                    out_trig[((long)b * L_ + row) * TCLS + col] = tacc[n][r] + bt[col];
            }
        }
    }

    // ---- span means (8 trigger spans + 16 candidate spans, H padded to 208)
    for (int i = tid; i < 24 * HP; i += 256) {
        int h = i % HP, s = i / HP;
        int st, ln; float* dst;
        if (s < NT_) {
            st = tstart[b * NT_ + s]; ln = tlen[b * NT_ + s];
            dst = trgS + ((long)b * NT_ + s) * HP + h;
        } else {
            int c = s - NT_;
            st = cstart[b * NC_ + c]; ln = clen[b * NC_ + c];
            dst = entS + ((long)b * NC_ + c) * HP + h;
        }
        int cnt = ln + 1;                       // span length >= 1, end <= 126
        float sum = 0.f;
        for (int r = 0; r < cnt; ++r) sum += cs[(st + r) * CS_STRIDE + h];
        *dst = sum / (float)cnt;
    }
}

// ---------------------------------------------------------------------------
// Kernel 2: argument GEMM. One block per batch (8 waves = the 8 triggers);
// M-tile = the 16 candidates of (b,t). The 24 span vectors of this batch and
// the whole packed wa are staged in LDS (126KB), so the K=416 inner loop is
// pure ds_load_b64 + WMMA. Span rows use stride 210 to avoid bank conflicts.
// ---------------------------------------------------------------------------
__global__ __launch_bounds__(256) void arg_kernel(
    const float* __restrict__ trgS, const float* __restrict__ entS,
    const float* __restrict__ wapG, const float* __restrict__ ba,
    float* __restrict__ out_arg) {
    extern __shared__ float asmem[];
    float* sp  = asmem;                 // [24][SSTR]: rows 0-7 = trg, 8-23 = ent
    float* swp = asmem + 24 * SSTR;     // packed wa, SZ_WAP floats

    const int b    = blockIdx.x;
    const int tid  = threadIdx.x;
    const int lane = tid & 31;
    const int wave = tid >> 5;          // = trigger index t
    const int col16 = lane & 15;
    const int koff  = (lane >> 4) << 1;
    const int rofs  = (lane >> 4) << 3;

    // stage spans (strided) and wa (flat)
    for (int i = tid; i < 24 * HP; i += 256) {
        int r = i / HP, h = i % HP;
        float v = (r < NT_) ? trgS[((long)b * NT_ + r) * HP + h]
                            : entS[((long)b * NC_ + (r - NT_)) * HP + h];
        sp[r * SSTR + h] = v;
    }
    for (int i = tid; i < SZ_WAP; i += 256) swp[i] = wapG[i];
    __syncthreads();

    const float* trow = sp + wave * SSTR;                  // broadcast across M
    const float* erow = sp + (NT_ + col16) * SSTR;         // A-row = candidate

    const v8f vzero = {0.f,0.f,0.f,0.f,0.f,0.f,0.f,0.f};
    v8f acc[4];
#pragma unroll
    for (int n = 0; n < 4; ++n) acc[n] = vzero;

#pragma unroll 2
    for (int kk = 0; kk < KKA; ++kk) {
        int k = (kk << 2) + koff;
        v2f a;
        if (k < HP) { a.x = trow[k];      a.y = trow[k + 1]; }      // uniform per kk
        else        { a.x = erow[k - HP]; a.y = erow[k - HP + 1]; }
        const float* pb = swp + ((kk * AP + col16) << 2) + koff;
#pragma unroll
        for (int n = 0; n < 4; ++n) {
            v2f bb = *(const v2f*)(pb + ((n * 16) << 2));
            acc[n] = wmma_f32(a, bb, acc[n]);
        }
    }
#pragma unroll
    for (int n = 0; n < 4; ++n) {
#pragma unroll
        for (int r = 0; r < 8; ++r) {
            int c   = r + rofs;               // candidate index = M row
            int col = n * 16 + col16;
            if (col < ACLS)
                out_arg[(((long)b * NT_ + wave) * NC_ + c) * ACLS + col] = acc[n][r] + ba[col];
        }
    }
}

// ---------------------------------------------------------------------------
extern "C" void kernel_launch(void* const* d_in, const int* in_sizes, int n_in,
                              void* d_out, int out_size, void* d_ws, size_t ws_size,
                              hipStream_t stream) {
    const int*   tokens = (const int*)d_in[0];
    const int*   cstart = (const int*)d_in[1];
    const int*   clen   = (const int*)d_in[2];
    const int*   tstart = (const int*)d_in[3];
    const int*   tlen   = (const int*)d_in[4];
    const float* emb    = (const float*)d_in[5];
    const float* w0     = (const float*)d_in[6];
    const float* w1     = (const float*)d_in[7];
    const float* w2     = (const float*)d_in[8];
    const float* w3     = (const float*)d_in[9];
    const float* b0     = (const float*)d_in[10];
    const float* b1     = (const float*)d_in[11];
    const float* b2     = (const float*)d_in[12];
    const float* b3     = (const float*)d_in[13];
    const float* wt     = (const float*)d_in[14];
    const float* bt     = (const float*)d_in[15];
    const float* wa     = (const float*)d_in[16];
    const float* ba     = (const float*)d_in[17];

    float* ws  = (float*)d_ws;
    float* out = (float*)d_out;
    float* out_trig = out;
    float* out_arg  = out + (long)B_ * L_ * TCLS;

    const int total_pack = SZ_WPACK + SZ_WTP + SZ_WAP + SZ_BIAS;
    pack_kernel<<<(total_pack + 255) / 256, 256, 0, stream>>>(
        w0, w1, w2, w3, b0, b1, b2, b3, wt, wa, ws);

    fused_kernel<<<B_, 256, SMEM_FUSED, stream>>>(
        tokens, cstart, clen, tstart, tlen, emb, bt, ws,
        ws + OFF_TRGS, ws + OFF_ENTS, out_trig);

    arg_kernel<<<B_, 256, SMEM_ARG, stream>>>(
        ws + OFF_TRGS, ws + OFF_ENTS, ws + OFF_WAP, ba, out_arg);
}